// SAGE_64226940944915
// MI455X (gfx1250) — compile-verified
//
#include <hip/hip_runtime.h>

typedef __attribute__((ext_vector_type(2))) float v2f;
typedef __attribute__((ext_vector_type(8))) float v8f;

#define DFEAT 64
#define LDW   66   // padded LDS row pitch: 264B -> consecutive rows 2 banks apart

// ---------------------------------------------------------------- zero scratch
__global__ void __launch_bounds__(256) sage_zero(float* __restrict__ p, long long n) {
    long long i = (long long)blockIdx.x * blockDim.x + threadIdx.x;
    long long stride = (long long)gridDim.x * blockDim.x;
    for (; i < n; i += stride) p[i] = 0.0f;
}

// ------------------------------------------------- edge scatter: agg += x[src]
// one wave32 per edge; each lane owns 2 of the 64 features (float2 slice).
__global__ void __launch_bounds__(256) sage_scatter(
    const float* __restrict__ x, const int* __restrict__ src,
    const int* __restrict__ dst, float* __restrict__ agg,
    float* __restrict__ deg, int nEdges)
{
    int gid  = blockIdx.x * 256 + threadIdx.x;
    int e    = gid >> 5;
    if (e >= nEdges) return;
    int lane = threadIdx.x & 31;

    int s = src[e];
    int d = dst[e];

    const float2 v = *(const float2*)(x + (size_t)s * DFEAT + lane * 2);
    float* ag = agg + (size_t)d * DFEAT + lane * 2;
    atomicAdd(ag + 0, v.x);          // global_atomic_add_f32 (L2-resident agg)
    atomicAdd(ag + 1, v.y);
    if (lane == 0) atomicAdd(deg + d, 1.0f);
}

// ----------------- fused mean + dual GEMM: out = x@Ws^T + (agg/deg)@Wn^T + b
// one wave per 16-row tile; 4x (16x16) f32 accumulators; K=128 via
// v_wmma_f32_16x16x4_f32. Weights staged in LDS (conflict-free pitch).
__global__ void __launch_bounds__(256) sage_gemm(
    const float* __restrict__ x,     const float* __restrict__ agg,
    const float* __restrict__ deg,   const float* __restrict__ Wself,
    const float* __restrict__ Wneigh,const float* __restrict__ bias,
    float* __restrict__ out, int nNodes)
{
    __shared__ float Wlds[2 * DFEAT * LDW];
    int tid = threadIdx.x;
    for (int i = tid; i < DFEAT * DFEAT; i += 256) {
        int r = i >> 6, c = i & 63;
        Wlds[r * LDW + c]               = Wself[i];
        Wlds[DFEAT * LDW + r * LDW + c] = Wneigh[i];
    }
    __syncthreads();

    int wave = tid >> 5;
    int lane = tid & 31;
    int rowBase = (blockIdx.x * 8 + wave) * 16;
    if (rowBase >= nNodes) return;        // wave-uniform: EXEC stays all-ones

    int half = lane >> 4;                 // 0: K=k,k+1   1: K=k+2,k+3
    int l16  = lane & 15;
    int arow = rowBase + l16;
    int arowc = arow < nNodes ? arow : nNodes - 1;   // clamp loads only

    float invdeg = 1.0f / fmaxf(deg[arowc], 1.0f);
    const float* xr = x   + (size_t)arowc * DFEAT + half * 2;
    const float* ar = agg + (size_t)arowc * DFEAT + half * 2;

    v8f acc0 = {}, acc1 = {}, acc2 = {}, acc3 = {};

    #pragma unroll
    for (int m = 0; m < 2; ++m) {
        const float* Wl = Wlds + m * DFEAT * LDW;
        const float* arp = m ? ar : xr;
        for (int k = 0; k < DFEAT; k += 4) {
            // A fragment (16x4 fp32): lane half selects K pair
            v2f a;
            a.x = arp[k];
            a.y = arp[k + 1];
            if (m) { a.x *= invdeg; a.y *= invdeg; }   // fused mean

            // B fragments (4x16 fp32): B[k',j] = W[j, k+k'] (transpose via index)
            int kk = k + half * 2;
            const float* w0 = Wl + ( 0 + l16) * LDW + kk;
            const float* w1 = Wl + (16 + l16) * LDW + kk;
            const float* w2 = Wl + (32 + l16) * LDW + kk;
            const float* w3 = Wl + (48 + l16) * LDW + kk;
            v2f b0; b0.x = w0[0]; b0.y = w0[1];
            v2f b1; b1.x = w1[0]; b1.y = w1[1];
            v2f b2; b2.x = w2[0]; b2.y = w2[1];
            v2f b3; b3.x = w3[0]; b3.y = w3[1];

            acc0 = __builtin_amdgcn_wmma_f32_16x16x4_f32(false, a, false, b0,
                                                         (short)0, acc0, false, false);
            acc1 = __builtin_amdgcn_wmma_f32_16x16x4_f32(false, a, false, b1,
                                                         (short)0, acc1, false, false);
            acc2 = __builtin_amdgcn_wmma_f32_16x16x4_f32(false, a, false, b2,
                                                         (short)0, acc2, false, false);
            acc3 = __builtin_amdgcn_wmma_f32_16x16x4_f32(false, a, false, b3,
                                                         (short)0, acc3, false, false);
        }
    }

    float bb0 = bias[ 0 + l16];
    float bb1 = bias[16 + l16];
    float bb2 = bias[32 + l16];
    float bb3 = bias[48 + l16];

    // C/D layout: VGPR v -> row M = v + 8*half, col N = l16 (+16 per tile)
    #pragma unroll
    for (int v = 0; v < 8; ++v) {
        int orow = rowBase + half * 8 + v;
        if (orow < nNodes) {
            float* orp = out + (size_t)orow * DFEAT;
            orp[ 0 + l16] = acc0[v] + bb0;
            orp[16 + l16] = acc1[v] + bb1;
            orp[32 + l16] = acc2[v] + bb2;
            orp[48 + l16] = acc3[v] + bb3;
        }
    }
}

// ---------------------------------------------------------------------- launch
extern "C" void kernel_launch(void* const* d_in, const int* in_sizes, int n_in,
                              void* d_out, int out_size, void* d_ws, size_t ws_size,
                              hipStream_t stream)
{
    const float* x      = (const float*)d_in[0];
    const int*   src    = (const int*)  d_in[1];
    const int*   dst    = (const int*)  d_in[2];
    const float* Wself  = (const float*)d_in[3];
    const float* Wneigh = (const float*)d_in[4];
    const float* bias   = (const float*)d_in[5];
    float*       out    = (float*)d_out;

    int nNodes = in_sizes[0] / DFEAT;   // 100000
    int nEdges = in_sizes[1];           // 1250000

    float* agg = (float*)d_ws;                       // [nNodes * 64]
    float* deg = agg + (size_t)nNodes * DFEAT;       // [nNodes]

    long long zn = (long long)nNodes * DFEAT + nNodes;
    sage_zero<<<2048, 256, 0, stream>>>(agg, zn);

    long long sthreads = (long long)nEdges * 32;
    int sblocks = (int)((sthreads + 255) / 256);
    sage_scatter<<<sblocks, 256, 0, stream>>>(x, src, dst, agg, deg, nEdges);

    int tiles   = (nNodes + 15) / 16;
    int gblocks = (tiles + 7) / 8;      // 8 waves (tiles) per 256-thread block
    sage_gemm<<<gblocks, 256, 0, stream>>>(x, agg, deg, Wself, Wneigh, bias,
                                           out, nNodes);
}